// InstanceGCN_11158325035312
// MI455X (gfx1250) — compile-verified
//
#include <hip/hip_runtime.h>

typedef float v2f __attribute__((ext_vector_type(2)));
typedef float v8f __attribute__((ext_vector_type(8)));

#define GT 128   // gemm block threads (4 waves, wave32)
#define GR 64    // rows per gemm block (16 rows per wave)

// Y[N,64] = act( alpha1 * X1@W1^T  (+ X2@W2^T)  + bscale*bias )
// W row-major [64(out),64(in)]. Uses V_WMMA_F32_16X16X4_F32 (exact f32).
__global__ __launch_bounds__(GT)
void fused_gemm64(const float* __restrict__ X1, const float* __restrict__ W1, float alpha1,
                  const float* __restrict__ X2, const float* __restrict__ W2,
                  const float* __restrict__ bias, float bscale,
                  float* __restrict__ Y, int N, int do_relu)
{
    __shared__ float sW1[64 * 64];
    __shared__ float sW2[64 * 64];
    __shared__ float sX [GR * 64];
    __shared__ float sB [64];

    const int tid  = threadIdx.x;
    const int lane = tid & 31;
    const int wave = tid >> 5;
    const int hf   = lane >> 4;     // lane half: selects K pair (A) / K rows (B)
    const int l16  = lane & 15;
    const int r0   = blockIdx.x * GR;

    for (int i = tid; i < 64 * 64 / 4; i += GT)
        ((float4*)sW1)[i] = ((const float4*)W1)[i];
    if (W2) {
        for (int i = tid; i < 64 * 64 / 4; i += GT)
            ((float4*)sW2)[i] = ((const float4*)W2)[i];
    }
    if (tid < 64) sB[tid] = bias[tid];
    for (int i = tid; i < GR * 64 / 4; i += GT) {
        int row = i >> 4;
        float4 v = make_float4(0.f, 0.f, 0.f, 0.f);
        if (r0 + row < N) v = ((const float4*)(X1 + (size_t)(r0 + row) * 64))[i & 15];
        ((float4*)sX)[i] = v;
    }
    __syncthreads();

    v8f acc[4];
#pragma unroll
    for (int nt = 0; nt < 4; ++nt)
#pragma unroll
        for (int j = 0; j < 8; ++j) acc[nt][j] = 0.0f;

    const int arow = wave * 16 + l16;

    // phase 1: X1 @ W1^T
#pragma unroll
    for (int k0 = 0; k0 < 64; k0 += 4) {
        v2f a = *(const v2f*)&sX[arow * 64 + k0 + 2 * hf];
#pragma unroll
        for (int nt = 0; nt < 4; ++nt) {
            v2f b = *(const v2f*)&sW1[(nt * 16 + l16) * 64 + k0 + 2 * hf];
            acc[nt] = __builtin_amdgcn_wmma_f32_16x16x4_f32(
                false, a, false, b, (short)0, acc[nt], false, false);
        }
    }
    if (alpha1 != 1.0f) {
#pragma unroll
        for (int nt = 0; nt < 4; ++nt)
#pragma unroll
            for (int j = 0; j < 8; ++j) acc[nt][j] *= alpha1;
    }

    // phase 2 (optional): += X2 @ W2^T  (reuse sX)
    if (X2) {
        __syncthreads();
        for (int i = tid; i < GR * 64 / 4; i += GT) {
            int row = i >> 4;
            float4 v = make_float4(0.f, 0.f, 0.f, 0.f);
            if (r0 + row < N) v = ((const float4*)(X2 + (size_t)(r0 + row) * 64))[i & 15];
            ((float4*)sX)[i] = v;
        }
        __syncthreads();
#pragma unroll
        for (int k0 = 0; k0 < 64; k0 += 4) {
            v2f a = *(const v2f*)&sX[arow * 64 + k0 + 2 * hf];
#pragma unroll
            for (int nt = 0; nt < 4; ++nt) {
                v2f b = *(const v2f*)&sW2[(nt * 16 + l16) * 64 + k0 + 2 * hf];
                acc[nt] = __builtin_amdgcn_wmma_f32_16x16x4_f32(
                    false, a, false, b, (short)0, acc[nt], false, false);
            }
        }
    }

    // epilogue: C/D layout -> VGPR j holds row j (lanes 0-15) / row j+8 (lanes 16-31)
#pragma unroll
    for (int nt = 0; nt < 4; ++nt) {
        int col = nt * 16 + l16;
        float bv = sB[col] * bscale;
#pragma unroll
        for (int j = 0; j < 8; ++j) {
            int row = r0 + wave * 16 + j + 8 * hf;
            if (row < N) {
                float v = acc[nt][j] + bv;
                if (do_relu) v = fmaxf(v, 0.0f);
                Y[(size_t)row * 64 + col] = v;
            }
        }
    }
}

// h[n,64] = relu(x[n,K] @ W[64,K]^T + b), tiny K (7/4/6)
__global__ __launch_bounds__(256)
void embed_kernel(const float* __restrict__ x, int K,
                  const float* __restrict__ W, const float* __restrict__ b,
                  float* __restrict__ h, int n)
{
    __shared__ float sW[64 * 8];
    __shared__ float sB[64];
    for (int i = threadIdx.x; i < 64 * K; i += 256) sW[i] = W[i];
    if (threadIdx.x < 64) sB[threadIdx.x] = b[threadIdx.x];
    __syncthreads();
    int idx  = blockIdx.x * 256 + threadIdx.x;
    int node = idx >> 6, o = idx & 63;
    if (node >= n) return;
    const float* xr = x + (size_t)node * K;
    const float* wr = sW + o * K;
    float s = sB[o];
    for (int k = 0; k < K; ++k) s += xr[k] * wr[k];
    h[(size_t)idx] = fmaxf(s, 0.0f);
}

// segment-max via u32 atomics: pooled >= 0 (post-ReLU) so IEEE bits are
// monotone as unsigned; agg pre-zeroed == DGL empty-segment fill of 0.
__global__ __launch_bounds__(256)
void scatter_max64(const float* __restrict__ pooled, const int* __restrict__ src,
                   const int* __restrict__ dst, unsigned* __restrict__ agg, int ne)
{
    int idx = blockIdx.x * 256 + threadIdx.x;
    int e   = idx >> 4;
    if (e >= ne) return;
    int q = idx & 15;
    int s = src[e], d = dst[e];
    float4 v = ((const float4*)(pooled + (size_t)s * 64))[q];
    unsigned* p = agg + (size_t)d * 64 + q * 4;
    atomicMax(p + 0, __float_as_uint(v.x));
    atomicMax(p + 1, __float_as_uint(v.y));
    atomicMax(p + 2, __float_as_uint(v.z));
    atomicMax(p + 3, __float_as_uint(v.w));
}

__global__ __launch_bounds__(256)
void add_inplace(float* __restrict__ a, const float* __restrict__ b, int n)
{
    int i = blockIdx.x * 256 + threadIdx.x;
    if (i < n) a[i] += b[i];
}

// out[n] = h[n,:] . W3 + b3
__global__ __launch_bounds__(256)
void final_dot(const float* __restrict__ h, const float* __restrict__ W3,
               const float* __restrict__ b3, float* __restrict__ out, int n)
{
    __shared__ float sw[64];
    if (threadIdx.x < 64) sw[threadIdx.x] = W3[threadIdx.x];
    __syncthreads();
    int node = blockIdx.x * 256 + threadIdx.x;
    if (node >= n) return;
    const float4* row = (const float4*)(h + (size_t)node * 64);
    float s = b3[0];
#pragma unroll
    for (int q = 0; q < 16; ++q) {
        float4 v = row[q];
        s += v.x * sw[4 * q] + v.y * sw[4 * q + 1] + v.z * sw[4 * q + 2] + v.w * sw[4 * q + 3];
    }
    out[node] = s;
}

extern "C" void kernel_launch(void* const* d_in, const int* in_sizes, int n_in,
                              void* d_out, int out_size, void* d_ws, size_t ws_size,
                              hipStream_t stream)
{
    const float* var_x = (const float*)d_in[0];
    const float* con_x = (const float*)d_in[1];
    const float* soc_x = (const float*)d_in[2];
    const int* v2c_src = (const int*)d_in[3];
    const int* v2c_dst = (const int*)d_in[4];
    const int* s2c_src = (const int*)d_in[5];
    const int* s2c_dst = (const int*)d_in[6];
    const int* c2v_src = (const int*)d_in[7];
    const int* c2v_dst = (const int*)d_in[8];
    const int* c2s_src = (const int*)d_in[9];
    const int* c2s_dst = (const int*)d_in[10];

    const int NV = in_sizes[0] / 7, NC = in_sizes[1] / 4, NS = in_sizes[2] / 6;
    const int EV = in_sizes[3], ES = in_sizes[5];

    struct ConvP { const float *Wp, *bp, *Wn, *Wself, *b; };
    const float *Wv, *bv, *Wc, *bc, *Ws, *bs, *W1, *b1, *W2, *b2, *W3, *b3;
    ConvP cf[2], cb[2];
#define P(i) ((const float*)d_in[(i)])
    if (n_in >= 43 && in_sizes[11] == 64 * 7) {
        // insertion-order recursive flatten of params dict
        Wv = P(11); bv = P(12); Wc = P(13); bc = P(14); Ws = P(15); bs = P(16);
        W1 = P(17); b1 = P(18); W2 = P(19); b2 = P(20); W3 = P(21); b3 = P(22);
        int i = 23;
        for (int l = 0; l < 2; ++l) {
            cf[l] = ConvP{P(i), P(i + 1), P(i + 2), P(i + 3), P(i + 4)}; i += 5;
            cb[l] = ConvP{P(i), P(i + 1), P(i + 2), P(i + 3), P(i + 4)}; i += 5;
        }
    } else {
        // sorted-key (jax tree) flatten: conv0b, conv0f, conv1b, conv1f, emb, out
        // conv inner sorted: Wn, Wp, Wself, b, bp
        int i = 11;
        for (int l = 0; l < 2; ++l) {
            cb[l] = ConvP{P(i + 1), P(i + 4), P(i + 0), P(i + 2), P(i + 3)}; i += 5;
            cf[l] = ConvP{P(i + 1), P(i + 4), P(i + 0), P(i + 2), P(i + 3)}; i += 5;
        }
        // emb sorted: Wc, Ws, Wv, bc, bs, bv
        Wc = P(i); Ws = P(i + 1); Wv = P(i + 2); bc = P(i + 3); bs = P(i + 4); bv = P(i + 5); i += 6;
        // out sorted: W1, W2, W3, b1, b2, b3
        W1 = P(i); W2 = P(i + 1); W3 = P(i + 2); b1 = P(i + 3); b2 = P(i + 4); b3 = P(i + 5);
    }
#undef P

    // workspace carve-out (~113 MB)
    char* wsb = (char*)d_ws;
    size_t off = 0;
    auto take = [&](size_t elems) -> float* {
        float* p = (float*)(wsb + off);
        off += ((elems * sizeof(float)) + 255) & ~(size_t)255;
        return p;
    };
    float* hv    = take((size_t)NV * 64);   // var features
    float* hc    = take((size_t)NC * 64);   // con features
    float* hs    = take((size_t)NS * 64);   // soc features
    float* poolA = take((size_t)NV * 64);   // pooledV / pooledC / MLP temp
    float* poolB = take((size_t)NS * 64);   // pooledS
    float* aggA  = take((size_t)NV * 64);   // segment-max accum (var- or con-sized)
    float* aggB  = take((size_t)NC * 64);   // segment-max accum (con- or soc-sized)
    (void)ws_size; (void)out_size;

    auto cdiv = [](int a, int b) { return (a + b - 1) / b; };

    // embeddings
    embed_kernel<<<cdiv(NV * 64, 256), 256, 0, stream>>>(var_x, 7, Wv, bv, hv, NV);
    embed_kernel<<<cdiv(NC * 64, 256), 256, 0, stream>>>(con_x, 4, Wc, bc, hc, NC);
    embed_kernel<<<cdiv(NS * 64, 256), 256, 0, stream>>>(soc_x, 6, Ws, bs, hs, NS);

    // forward pass: update constraints (v2c + s2c share module params)
    for (int l = 0; l < 2; ++l) {
        const ConvP& p = cf[l];
        fused_gemm64<<<cdiv(NV, GR), GT, 0, stream>>>(hv, p.Wp, 1.0f, nullptr, nullptr, p.bp, 1.0f, poolA, NV, 1);
        fused_gemm64<<<cdiv(NS, GR), GT, 0, stream>>>(hs, p.Wp, 1.0f, nullptr, nullptr, p.bp, 1.0f, poolB, NS, 1);
        hipMemsetAsync(aggA, 0, (size_t)NC * 64 * 4, stream);
        hipMemsetAsync(aggB, 0, (size_t)NC * 64 * 4, stream);
        scatter_max64<<<cdiv(EV * 16, 256), 256, 0, stream>>>(poolA, v2c_src, v2c_dst, (unsigned*)aggA, EV);
        scatter_max64<<<cdiv(ES * 16, 256), 256, 0, stream>>>(poolB, s2c_src, s2c_dst, (unsigned*)aggB, ES);
        add_inplace<<<cdiv(NC * 64, 256), 256, 0, stream>>>(aggA, aggB, NC * 64);
        // h_con = relu( 2*h_con@Wself^T + (aggV+aggS)@Wn^T + 2*b )  (in-place safe)
        fused_gemm64<<<cdiv(NC, GR), GT, 0, stream>>>(hc, p.Wself, 2.0f, aggA, p.Wn, p.b, 2.0f, hc, NC, 1);
    }

    // backward pass: update vars + socs (c2v + c2s share module params -> one pooledC)
    for (int l = 0; l < 2; ++l) {
        const ConvP& p = cb[l];
        fused_gemm64<<<cdiv(NC, GR), GT, 0, stream>>>(hc, p.Wp, 1.0f, nullptr, nullptr, p.bp, 1.0f, poolA, NC, 1);
        hipMemsetAsync(aggA, 0, (size_t)NV * 64 * 4, stream);
        hipMemsetAsync(aggB, 0, (size_t)NS * 64 * 4, stream);
        scatter_max64<<<cdiv(EV * 16, 256), 256, 0, stream>>>(poolA, c2v_src, c2v_dst, (unsigned*)aggA, EV);
        scatter_max64<<<cdiv(ES * 16, 256), 256, 0, stream>>>(poolA, c2s_src, c2s_dst, (unsigned*)aggB, ES);
        fused_gemm64<<<cdiv(NV, GR), GT, 0, stream>>>(hv, p.Wself, 1.0f, aggA, p.Wn, p.b, 1.0f, hv, NV, 1);
        fused_gemm64<<<cdiv(NS, GR), GT, 0, stream>>>(hs, p.Wself, 1.0f, aggB, p.Wn, p.b, 1.0f, hs, NS, 1);
    }

    // output MLP on vars
    fused_gemm64<<<cdiv(NV, GR), GT, 0, stream>>>(hv, W1, 1.0f, nullptr, nullptr, b1, 1.0f, poolA, NV, 1);
    fused_gemm64<<<cdiv(NV, GR), GT, 0, stream>>>(poolA, W2, 1.0f, nullptr, nullptr, b2, 1.0f, poolA, NV, 1);
    final_dot<<<cdiv(NV, 256), 256, 0, stream>>>(poolA, W3, b3, (float*)d_out, NV);
}